// CombinedLoss_50843822850144
// MI455X (gfx1250) — compile-verified
//
#include <hip/hip_runtime.h>
#include <math.h>

// ---------------- problem constants (from reference) ----------------
#define NUM_CLASSES 4101
#define NPART 3
#define NB 2048          // batch
#define ND 2048          // feature dim
#define EPS 0.1f
#define MARGIN 0.3f

typedef float v2f __attribute__((ext_vector_type(2)));
typedef float v8f __attribute__((ext_vector_type(8)));

// Monotone uint key for float atomic max/min (handles +/-inf sentinels).
__device__ __forceinline__ unsigned fkey(float f) {
  unsigned b = __float_as_uint(f);
  return (b & 0x80000000u) ? ~b : (b | 0x80000000u);
}
__device__ __forceinline__ float fkey_inv(unsigned k) {
  return (k & 0x80000000u) ? __uint_as_float(k ^ 0x80000000u)
                           : __uint_as_float(~k);
}

// ---------------- workspace init ----------------
__global__ void init_ws(float* ce_acc, unsigned* __restrict__ apk,
                        unsigned* __restrict__ ank) {
  int i = blockIdx.x * blockDim.x + threadIdx.x;
  if (i == 0) *ce_acc = 0.0f;
  if (i < NB) {
    apk[i] = 0u;           // key(-NaN): below every real key -> max identity
    ank[i] = 0xFFFFFFFFu;  // key(+NaN): above every real key -> min identity
  }
}

// ---------------- label-smoothed CE (online softmax, one pass) ----------------
// loss_row = -((1-eps)*(x[t]-lse) + (eps/C)*(sum_x - C*lse))
__global__ __launch_bounds__(256) void ce_kernel(
    const float* __restrict__ logits, const int* __restrict__ targets,
    float* __restrict__ ce_acc) {
  const int b = blockIdx.x;
  const int p = blockIdx.y;
  const float* row = logits + ((size_t)p * NB + b) * NUM_CLASSES;
  const int tid = threadIdx.x;

  float mx = -__builtin_huge_valf();
  float s = 0.0f, sx = 0.0f;
  for (int c = tid; c < NUM_CLASSES; c += 256) {
    float x = row[c];
    sx += x;
    if (x > mx) {
      s = s * __expf(mx - x) + 1.0f;  // first iter: 0*exp(-inf)=0 -> s=1
      mx = x;
    } else {
      s += __expf(x - mx);
    }
  }

  __shared__ float sm[256], ss[256], sxa[256];
  sm[tid] = mx; ss[tid] = s; sxa[tid] = sx;
  __syncthreads();
  for (int off = 128; off > 0; off >>= 1) {
    if (tid < off) {
      float m2 = sm[tid + off], s2 = ss[tid + off];
      float M = fmaxf(sm[tid], m2);
      ss[tid] = ss[tid] * __expf(sm[tid] - M) + s2 * __expf(m2 - M);
      sm[tid] = M;
      sxa[tid] += sxa[tid + off];
    }
    __syncthreads();
  }
  if (tid == 0) {
    float lse = sm[0] + __logf(ss[0]);
    float xt = row[targets[b]];
    float loss = -((1.0f - EPS) * (xt - lse) +
                   (EPS / (float)NUM_CLASSES) * (sxa[0] - (float)NUM_CLASSES * lse));
    atomicAdd(ce_acc, loss * (1.0f / (float)(NPART * NB)));
  }
}

// ---------------- per-row squared norms ----------------
__global__ __launch_bounds__(256) void sq_kernel(const float* __restrict__ G,
                                                 float* __restrict__ sq) {
  const int i = blockIdx.x;
  const float* row = G + (size_t)i * ND;
  float s = 0.0f;
  for (int c = threadIdx.x; c < ND; c += 256) {
    float x = row[c];
    s += x * x;
  }
  __shared__ float sh[256];
  sh[threadIdx.x] = s;
  __syncthreads();
  for (int off = 128; off > 0; off >>= 1) {
    if (threadIdx.x < off) sh[threadIdx.x] += sh[threadIdx.x + off];
    __syncthreads();
  }
  if (threadIdx.x == 0) sq[i] = sh[0];
}

// ---------------- fused G@G^T (fp32 WMMA) + distance + batch-hard mining ------
// Block = 256 threads (8 wave32s) computes a 64x64 tile of D = G G^T.
// Wave w owns sub-tiles (r = w&3, c = 2*(w>>2)) and (r, c+1); each is a 16x16
// V_WMMA_F32_16X16X4_F32 accumulation over K=2048 staged through LDS.
#define KC 32
#define LSTR 36  // LDS row stride in floats: 36 mod 64 -> 16 distinct banks, 16B-aligned rows
__global__ __launch_bounds__(256) void dist_tile(
    const float* __restrict__ G, const int* __restrict__ tg,
    const float* __restrict__ sq, unsigned* __restrict__ apk,
    unsigned* __restrict__ ank) {
  __shared__ float As[64 * LSTR];
  __shared__ float Bs[64 * LSTR];

  const int i0 = blockIdx.y * 64;
  const int j0 = blockIdx.x * 64;
  const int t = threadIdx.x;
  const int lane = t & 31;
  const int w = t >> 5;
  const int m = lane & 15;   // N-lane within 16
  const int hi = lane >> 4;  // half-wave select
  const int r = w & 3;       // sub-tile row
  const int cb = (w >> 2) * 2;  // first of two sub-tile cols

  // cooperative tile load mapping: 64 rows x 32 K-cols, 8 floats/thread
  const int lrow = t >> 2;
  const int lcg = (t & 3) * 8;
  const float* gA = G + (size_t)(i0 + lrow) * ND + lcg;
  const float* gB = G + (size_t)(j0 + lrow) * ND + lcg;
  float* sA = &As[lrow * LSTR + lcg];
  float* sB = &Bs[lrow * LSTR + lcg];

  v8f acc0 = {};
  v8f acc1 = {};

  for (int k0 = 0; k0 < ND; k0 += KC) {
    float4 a0 = *(const float4*)(gA + k0);
    float4 a1 = *(const float4*)(gA + k0 + 4);
    float4 b0 = *(const float4*)(gB + k0);
    float4 b1 = *(const float4*)(gB + k0 + 4);
    if (k0 + KC < ND) {
      __builtin_prefetch(gA + k0 + KC, 0, 3);  // global_prefetch_b8
      __builtin_prefetch(gB + k0 + KC, 0, 3);
    }
    *(float4*)(sA) = a0;
    *(float4*)(sA + 4) = a1;
    *(float4*)(sB) = b0;
    *(float4*)(sB + 4) = b1;
    __syncthreads();

    // A frag layout: lane m holds M=m; K pair {kk,kk+1} (lanes 0-15) or {kk+2,kk+3}
    const float* arow = &As[(r * 16 + m) * LSTR + 2 * hi];
    const float* brow0 = &Bs[((cb + 0) * 16 + m) * LSTR + 2 * hi];
    const float* brow1 = &Bs[((cb + 1) * 16 + m) * LSTR + 2 * hi];
#pragma unroll
    for (int kk = 0; kk < KC; kk += 4) {
      v2f a = *(const v2f*)(arow + kk);
      v2f bv0 = *(const v2f*)(brow0 + kk);
      v2f bv1 = *(const v2f*)(brow1 + kk);
      acc0 = __builtin_amdgcn_wmma_f32_16x16x4_f32(false, a, false, bv0,
                                                   (short)0, acc0, false, false);
      acc1 = __builtin_amdgcn_wmma_f32_16x16x4_f32(false, a, false, bv1,
                                                   (short)0, acc1, false, false);
    }
    __syncthreads();
  }

  // Epilogue: dist + mask + hardest-pos/neg mining, fused (no 16MB dist matrix).
  const float INF = __builtin_huge_valf();
  const int jc0 = j0 + (cb + 0) * 16 + m;
  const int jc1 = j0 + (cb + 1) * 16 + m;
  const float sqj0 = sq[jc0], sqj1 = sq[jc1];
  const int tj0 = tg[jc0], tj1 = tg[jc1];
#pragma unroll
  for (int v = 0; v < 8; ++v) {
    const int i = i0 + r * 16 + v + hi * 8;  // C/D layout: M = v + 8*hi
    const float sqi = sq[i];
    const int ti = tg[i];

    float d0 = sqrtf(fmaxf(sqi + sqj0 - 2.0f * acc0[v], 1e-12f));
    float d1 = sqrtf(fmaxf(sqi + sqj1 - 2.0f * acc1[v], 1e-12f));
    float ap = fmaxf((ti == tj0) ? d0 : -INF, (ti == tj1) ? d1 : -INF);
    float an = fminf((ti == tj0) ? INF : d0, (ti == tj1) ? INF : d1);

#pragma unroll
    for (int msk = 1; msk < 16; msk <<= 1) {  // reduce across the 16 N-lanes
      ap = fmaxf(ap, __shfl_xor(ap, msk, 32));
      an = fminf(an, __shfl_xor(an, msk, 32));
    }
    if (m == 0) {
      atomicMax(&apk[i], fkey(ap));
      atomicMin(&ank[i], fkey(an));
    }
  }
}

// ---------------- finalize: tri = mean(relu(ap - an + margin)); out = ce + tri
__global__ __launch_bounds__(256) void final_kernel(
    const float* __restrict__ ce_acc, const unsigned* __restrict__ apk,
    const unsigned* __restrict__ ank, float* __restrict__ out) {
  float s = 0.0f;
  for (int i = threadIdx.x; i < NB; i += 256) {
    float ap = fkey_inv(apk[i]);
    float an = fkey_inv(ank[i]);
    s += fmaxf(ap - an + MARGIN, 0.0f);
  }
  __shared__ float sh[256];
  sh[threadIdx.x] = s;
  __syncthreads();
  for (int off = 128; off > 0; off >>= 1) {
    if (threadIdx.x < off) sh[threadIdx.x] += sh[threadIdx.x + off];
    __syncthreads();
  }
  if (threadIdx.x == 0) out[0] = *ce_acc + sh[0] * (1.0f / (float)NB);
}

// ---------------- launcher ----------------
extern "C" void kernel_launch(void* const* d_in, const int* in_sizes, int n_in,
                              void* d_out, int out_size, void* d_ws,
                              size_t ws_size, hipStream_t stream) {
  const float* part_logits = (const float*)d_in[0];  // [3,2048,4101] f32
  const float* gfeat = (const float*)d_in[1];        // [2048,2048] f32
  const int* targets = (const int*)d_in[2];          // [2048] i32
  float* out = (float*)d_out;

  // workspace layout (~25 KB)
  float* ce_acc = (float*)d_ws;                    // 1 float @ 0
  unsigned* apk = (unsigned*)d_ws + 64;            // 2048 keys
  unsigned* ank = apk + NB;                        // 2048 keys
  float* sq = (float*)(ank + NB);                  // 2048 floats

  init_ws<<<dim3((NB + 255) / 256), 256, 0, stream>>>(ce_acc, apk, ank);
  ce_kernel<<<dim3(NB, NPART), 256, 0, stream>>>(part_logits, targets, ce_acc);
  sq_kernel<<<dim3(NB), 256, 0, stream>>>(gfeat, sq);
  dist_tile<<<dim3(NB / 64, NB / 64), 256, 0, stream>>>(gfeat, targets, sq,
                                                        apk, ank);
  final_kernel<<<1, 256, 0, stream>>>(ce_acc, apk, ank, out);
}